// Early_fuse_65274912965115
// MI455X (gfx1250) — compile-verified
//
#include <hip/hip_runtime.h>
#include <hip/hip_bf16.h>

// ---------------------------------------------------------------------------
// Problem constants (from reference)
// ---------------------------------------------------------------------------
#define N_      256
#define C_      128
#define H_      77
#define W_      44
#define P_      7
#define MAXH_   11
#define POSEC_  256
#define V_      17
#define OC_     512
#define FUSE_   1920          // C_*MAXH_ + POSEC_*2 = 1408 + 512
#define SILF_   1408          // C_*MAXH_
#define TOK_    1792          // N_*P_
#define HEADS_  8
#define DHEAD_  64
#define FFH_    1024
#define DEPTH_  4
#define EPS_    1e-5f

typedef __attribute__((ext_vector_type(2))) float v2f;
typedef __attribute__((ext_vector_type(8))) float v8f;

#if __has_builtin(__builtin_amdgcn_wmma_f32_16x16x4_f32)
#define USE_WMMA_F32 1
#endif

__device__ __forceinline__ float gelu_exact(float x) {
    return 0.5f * x * (1.0f + erff(x * 0.70710678118654752440f));
}

// ---------------------------------------------------------------------------
// K1: silhouette segment pool.  fuse_raw[(n*P+p)*FUSE + c*11+h] =
//     mean_w sil[n,c,start+h,w] + max_w sil[n,c,start+h,w]
// Each thread reduces one W-row (44 floats, 16B-aligned -> 11x float4).
// This kernel carries the HBM floor of the whole op (~444 MB of sil reads).
// ---------------------------------------------------------------------------
__global__ void sil_pool_k(const float* __restrict__ sil,
                           const int* __restrict__ mins,
                           float* __restrict__ fuse_raw) {
    int tid = blockIdx.x * blockDim.x + threadIdx.x;
    const int total = N_ * P_ * SILF_;
    if (tid >= total) return;
    int f = tid % SILF_;
    int p = (tid / SILF_) % P_;
    int n = tid / (SILF_ * P_);
    int c = f / MAXH_;
    int h = f % MAXH_;
    int start = mins[p * N_ + n];
    start = start < 0 ? 0 : (start > (H_ - MAXH_) ? (H_ - MAXH_) : start);
    const float4* row = (const float4*)(sil +
        (((size_t)n * C_ + c) * H_ + (size_t)(start + h)) * W_);
    float sum = 0.f;
    float mx  = -3.4e38f;
#pragma unroll
    for (int q = 0; q < W_ / 4; ++q) {
        float4 v = row[q];
        sum += v.x + v.y + v.z + v.w;
        float m1 = fmaxf(fmaxf(v.x, v.y), fmaxf(v.z, v.w));
        mx = fmaxf(mx, m1);
    }
    fuse_raw[((size_t)n * P_ + p) * FUSE_ + f] = sum * (1.0f / (float)W_) + mx;
}

// ---------------------------------------------------------------------------
// K2: pose bins + leaky_relu(0.01).
// group 0 = joints {0..4}: bin0 = mean(j0,j1,j2), bin1 = mean(j2,j3,j4)
// group p>=1 = joints {2p+3, 2p+4}: bin i = joint (2p+3+i)
// fuse_raw[(n*P+p)*FUSE + 1408 + c*2 + i]
// ---------------------------------------------------------------------------
__global__ void pose_k(const float* __restrict__ pose,
                       float* __restrict__ fuse_raw) {
    int tid = blockIdx.x * blockDim.x + threadIdx.x;
    const int total = N_ * P_ * POSEC_ * 2;
    if (tid >= total) return;
    int f2 = tid % (POSEC_ * 2);
    int p  = (tid / (POSEC_ * 2)) % P_;
    int n  = tid / (POSEC_ * 2 * P_);
    int c  = f2 >> 1;
    int i  = f2 & 1;
    const float* pb = pose + ((size_t)n * POSEC_ + c) * V_;
    float val;
    if (p == 0) {
        val = (i == 0) ? (pb[0] + pb[1] + pb[2]) * (1.0f / 3.0f)
                       : (pb[2] + pb[3] + pb[4]) * (1.0f / 3.0f);
    } else {
        val = pb[2 * p + 3 + i];
    }
    val = (val > 0.f) ? val : 0.01f * val;
    fuse_raw[((size_t)n * P_ + p) * FUSE_ + SILF_ + f2] = val;
}

// ---------------------------------------------------------------------------
// K3: row LayerNorm:  y = (x - mean) * rsqrt(var + eps) * g + b
// one block (256 thr = 8 waves) per row.
// ---------------------------------------------------------------------------
__global__ __launch_bounds__(256) void ln_rows_k(const float* __restrict__ x,
                                                 const float* __restrict__ g,
                                                 const float* __restrict__ b,
                                                 float* __restrict__ y, int D) {
    __shared__ float s1[256];
    __shared__ float s2[256];
    const int row = blockIdx.x;
    const float* xr = x + (size_t)row * D;
    float* yr = y + (size_t)row * D;
    float sum = 0.f, sq = 0.f;
    for (int d = threadIdx.x; d < D; d += 256) {
        float v = xr[d];
        sum += v;
        sq  += v * v;
    }
    s1[threadIdx.x] = sum;
    s2[threadIdx.x] = sq;
    __syncthreads();
    for (int o = 128; o > 0; o >>= 1) {
        if ((int)threadIdx.x < o) {
            s1[threadIdx.x] += s1[threadIdx.x + o];
            s2[threadIdx.x] += s2[threadIdx.x + o];
        }
        __syncthreads();
    }
    float mean = s1[0] / (float)D;
    float var  = s2[0] / (float)D - mean * mean;
    float inv  = rsqrtf(var + EPS_);
    for (int d = threadIdx.x; d < D; d += 256) {
        yr[d] = (xr[d] - mean) * inv * g[d] + b[d];
    }
}

// ---------------------------------------------------------------------------
// K4: fp32 WMMA GEMM, register-blocked.
//   C[M,N] = act(A[M,K] @ W[K,N] + bias) (+ residual)
// Block = 128 threads = 4 waves covering a 128x64 output macro-tile
// (waves in 2x2: wave>>1 -> M half, wave&1 -> N half).
// Each wave computes 64x32 = 4 M-tiles x 2 N-tiles of 16x16, i.e. 8 v8f
// accumulators.  Per K-step of 4: 4 contiguous b64 A-fragment loads +
// 2x2 strided b32 B loads feed 8 independent v_wmma_f32_16x16x4_f32
// (8 vmem per 8 wmma; independent WMMAs hide XDL latency).
// V_WMMA_F32_16X16X4_F32 fragment layout (ISA 7.12.2):
//   A 16x4 : lane&15 = M row, lane>>4 selects K{0,1}|{2,3}, VGPR v -> K low bit
//   B 4x16 : lane&15 = N col, VGPR v holds K=v (lanes0-15) / K=v+2 (lanes16-31)
//   C/D    : VGPR j holds row +j (lanes0-15) / +j+8 (lanes16-31)
// M=1792=14*128, N in {512,1024,1536} all /64, K /4 -> exact tiling, EXEC
// stays all-ones as WMMA requires.
// ---------------------------------------------------------------------------
__global__ __launch_bounds__(128) void gemm_wmma_k(
    const float* __restrict__ A, const float* __restrict__ Wt,
    const float* __restrict__ bias, const float* __restrict__ res,
    float* __restrict__ Cm, int K, int Nout, int act) {
    const int lane  = threadIdx.x & 31;
    const int l15   = lane & 15;
    const int lhalf = lane >> 4;
    const int wave  = threadIdx.x >> 5;
    const int m0    = blockIdx.y * 128 + (wave >> 1) * 64;
    const int n0    = blockIdx.x * 64 + (wave & 1) * 32;

    const float* arow0 = A + (size_t)(m0 + l15) * K + 2 * lhalf;
    const float* wcol0 = Wt + (size_t)(2 * lhalf) * Nout + n0 + l15;

    v8f acc[4][2];
#pragma unroll
    for (int mt = 0; mt < 4; ++mt)
#pragma unroll
        for (int nt = 0; nt < 2; ++nt)
            acc[mt][nt] = (v8f){0.f, 0.f, 0.f, 0.f, 0.f, 0.f, 0.f, 0.f};

#if defined(USE_WMMA_F32)
    for (int k = 0; k < K; k += 4) {
        if ((k & 255) == 0 && (k + 256) < K)
            __builtin_prefetch(wcol0 + (size_t)(k + 256) * Nout, 0, 0);
        v2f af[4];
        v2f bf[2];
#pragma unroll
        for (int mt = 0; mt < 4; ++mt) {
            const float* ap = arow0 + (size_t)mt * 16 * K + k;
            af[mt][0] = ap[0];
            af[mt][1] = ap[1];
        }
#pragma unroll
        for (int nt = 0; nt < 2; ++nt) {
            const float* bp = wcol0 + (size_t)k * Nout + nt * 16;
            bf[nt][0] = bp[0];
            bf[nt][1] = bp[Nout];
        }
#pragma unroll
        for (int mt = 0; mt < 4; ++mt)
#pragma unroll
            for (int nt = 0; nt < 2; ++nt)
                acc[mt][nt] = __builtin_amdgcn_wmma_f32_16x16x4_f32(
                    /*neg_a=*/false, af[mt], /*neg_b=*/false, bf[nt],
                    /*c_mod=*/(short)0, acc[mt][nt],
                    /*reuse_a=*/false, /*reuse_b=*/false);
    }
#else
    // scalar fallback (kept only so the file always compiles)
    for (int k = 0; k < K; ++k) {
#pragma unroll
        for (int nt = 0; nt < 2; ++nt) {
            float wv = Wt[(size_t)k * Nout + n0 + nt * 16 + l15];
#pragma unroll
            for (int mt = 0; mt < 4; ++mt)
#pragma unroll
                for (int j = 0; j < 8; ++j)
                    acc[mt][nt][j] +=
                        A[(size_t)(m0 + mt * 16 + j + 8 * lhalf) * K + k] * wv;
        }
    }
#endif

    float bv[2];
    bv[0] = bias ? bias[n0 + l15] : 0.f;
    bv[1] = bias ? bias[n0 + 16 + l15] : 0.f;
#pragma unroll
    for (int mt = 0; mt < 4; ++mt) {
#pragma unroll
        for (int nt = 0; nt < 2; ++nt) {
#pragma unroll
            for (int j = 0; j < 8; ++j) {
                int row = m0 + mt * 16 + j + 8 * lhalf;
                size_t idx = (size_t)row * Nout + n0 + nt * 16 + l15;
                float v = acc[mt][nt][j] + bv[nt];
                if (act == 1) v = gelu_exact(v);
                if (res) v += res[idx];
                Cm[idx] = v;
            }
        }
    }
}

// ---------------------------------------------------------------------------
// K5: attention for one (n, head): q,k,v are 7x64; softmax(q k^T / 8) v
// ---------------------------------------------------------------------------
__global__ __launch_bounds__(64) void attn_k(const float* __restrict__ qkv,
                                             float* __restrict__ obuf) {
    __shared__ float q[P_][DHEAD_];
    __shared__ float kk[P_][DHEAD_];
    __shared__ float vv[P_][DHEAD_];
    __shared__ float at[P_][P_];
    const int n = blockIdx.x / HEADS_;
    const int h = blockIdx.x % HEADS_;
    const float* base = qkv + (size_t)n * P_ * (3 * OC_) + h * DHEAD_;
    for (int idx = threadIdx.x; idx < P_ * DHEAD_; idx += 64) {
        int i = idx / DHEAD_, d = idx % DHEAD_;
        q[i][d]  = base[(size_t)i * (3 * OC_) + d];
        kk[i][d] = base[(size_t)i * (3 * OC_) + OC_ + d];
        vv[i][d] = base[(size_t)i * (3 * OC_) + 2 * OC_ + d];
    }
    __syncthreads();
    if ((int)threadIdx.x < P_ * P_) {
        int i = threadIdx.x / P_, j = threadIdx.x % P_;
        float s = 0.f;
#pragma unroll
        for (int d = 0; d < DHEAD_; ++d) s += q[i][d] * kk[j][d];
        at[i][j] = s * 0.125f;  // DHEAD^-0.5 = 1/8
    }
    __syncthreads();
    if ((int)threadIdx.x < P_) {
        int i = threadIdx.x;
        float m = at[i][0];
#pragma unroll
        for (int j = 1; j < P_; ++j) m = fmaxf(m, at[i][j]);
        float s = 0.f;
#pragma unroll
        for (int j = 0; j < P_; ++j) {
            float e = expf(at[i][j] - m);
            at[i][j] = e;
            s += e;
        }
        float inv = 1.0f / s;
#pragma unroll
        for (int j = 0; j < P_; ++j) at[i][j] *= inv;
    }
    __syncthreads();
    for (int idx = threadIdx.x; idx < P_ * DHEAD_; idx += 64) {
        int i = idx / DHEAD_, d = idx % DHEAD_;
        float s = 0.f;
#pragma unroll
        for (int j = 0; j < P_; ++j) s += at[i][j] * vv[j][d];
        obuf[((size_t)n * P_ + i) * OC_ + h * DHEAD_ + d] = s;
    }
}

// ---------------------------------------------------------------------------
// host orchestration
// ---------------------------------------------------------------------------
static inline void launch_gemm(const float* A, const float* Wt, const float* bias,
                               const float* res, float* Cm, int K, int Nout,
                               int act, hipStream_t stream) {
    dim3 grid(Nout / 64, TOK_ / 128);  // 4 waves/block, 128x64 tile per block
    gemm_wmma_k<<<grid, 128, 0, stream>>>(A, Wt, bias, res, Cm, K, Nout, act);
}

extern "C" void kernel_launch(void* const* d_in, const int* in_sizes, int n_in,
                              void* d_out, int out_size, void* d_ws, size_t ws_size,
                              hipStream_t stream) {
    const float* sil   = (const float*)d_in[0];
    const float* pose  = (const float*)d_in[1];
    const int*   mins  = (const int*)d_in[2];
    // d_in[3] = maxs (unused, mins+MAX_H by construction)
    const float* ln0_g = (const float*)d_in[4];
    const float* ln0_b = (const float*)d_in[5];
    const float* We    = (const float*)d_in[6];
    const float* be    = (const float*)d_in[7];
    const float* ln1_g = (const float*)d_in[8];
    const float* ln1_b = (const float*)d_in[9];
    const float* lnag  = (const float*)d_in[10];
    const float* lnab  = (const float*)d_in[11];
    const float* Wqkv  = (const float*)d_in[12];
    const float* Wo    = (const float*)d_in[13];
    const float* bo    = (const float*)d_in[14];
    const float* lnfg  = (const float*)d_in[15];
    const float* lnfb  = (const float*)d_in[16];
    const float* W1    = (const float*)d_in[17];
    const float* b1    = (const float*)d_in[18];
    const float* W2    = (const float*)d_in[19];
    const float* b2    = (const float*)d_in[20];
    float* out = (float*)d_out;

    // workspace arena (floats), with temporal reuse:
    //  region1 [0, 3440640)          : fuse_raw  -> qkv
    //  region2 [3440640, 6881280)    : a_ln      -> xbuf -> obuf -> ybuf/hbuf
    //  sA, sB  (residual stream ping-pong)
    float* wsf      = (float*)d_ws;
    float* fuse_raw = wsf;
    float* qkvb     = wsf;
    float* a_ln     = wsf + (size_t)3440640;
    float* xbuf     = a_ln;
    float* obuf     = a_ln;
    float* ybuf     = a_ln;
    float* hbuf     = a_ln + (size_t)917504;
    float* sA       = wsf + (size_t)6881280;
    float* sB       = sA + (size_t)917504;

    // ---- stage 1: pooling + pose -> fuse_raw (1792 x 1920) ----
    {
        int total = N_ * P_ * SILF_;
        sil_pool_k<<<(total + 255) / 256, 256, 0, stream>>>(sil, mins, fuse_raw);
        total = N_ * P_ * POSEC_ * 2;
        pose_k<<<(total + 255) / 256, 256, 0, stream>>>(pose, fuse_raw);
    }

    // ---- stage 2: LN0, embed GEMM (1920 -> 512), LN1 ----
    ln_rows_k<<<TOK_, 256, 0, stream>>>(fuse_raw, ln0_g, ln0_b, a_ln, FUSE_);
    launch_gemm(a_ln, We, be, nullptr, sB, FUSE_, OC_, 0, stream);
    ln_rows_k<<<TOK_, 256, 0, stream>>>(sB, ln1_g, ln1_b, sA, OC_);

    float* cur = sA;
    float* nxt = sB;

    // ---- stage 3: 4 transformer layers ----
    for (int l = 0; l < DEPTH_; ++l) {
        const float* Wqkv_l = Wqkv + (size_t)l * OC_ * (3 * OC_);
        const float* Wo_l   = Wo   + (size_t)l * OC_ * OC_;
        const float* bo_l   = bo   + (size_t)l * OC_;
        const float* W1_l   = W1   + (size_t)l * OC_ * FFH_;
        const float* b1_l   = b1   + (size_t)l * FFH_;
        const float* W2_l   = W2   + (size_t)l * FFH_ * OC_;
        const float* b2_l   = b2   + (size_t)l * OC_;

        // attention block
        ln_rows_k<<<TOK_, 256, 0, stream>>>(cur, lnag + (size_t)l * OC_,
                                            lnab + (size_t)l * OC_, xbuf, OC_);
        launch_gemm(xbuf, Wqkv_l, nullptr, nullptr, qkvb, OC_, 3 * OC_, 0, stream);
        attn_k<<<N_ * HEADS_, 64, 0, stream>>>(qkvb, obuf);
        launch_gemm(obuf, Wo_l, bo_l, cur, nxt, OC_, OC_, 0, stream);
        { float* t = cur; cur = nxt; nxt = t; }

        // feed-forward block
        ln_rows_k<<<TOK_, 256, 0, stream>>>(cur, lnfg + (size_t)l * OC_,
                                            lnfb + (size_t)l * OC_, ybuf, OC_);
        launch_gemm(ybuf, W1_l, b1_l, nullptr, hbuf, OC_, FFH_, 1, stream);
        float* dst = (l == DEPTH_ - 1) ? out : nxt;
        launch_gemm(hbuf, W2_l, b2_l, cur, dst, FFH_, OC_, 0, stream);
        if (l != DEPTH_ - 1) { float* t = cur; cur = nxt; nxt = t; }
    }
    (void)in_sizes; (void)n_in; (void)out_size; (void)ws_size;
}